// HBiLSTM_19404662243372
// MI455X (gfx1250) — compile-verified
//
#include <hip/hip_runtime.h>
#include <cstddef>

#define B_   32
#define T_   512
#define DIN_ 512
#define H_   256

typedef float v2f __attribute__((ext_vector_type(2)));
typedef float v8f __attribute__((ext_vector_type(8)));

// D = A(16x4 f32) * B(4x16 f32) + C(16x16 f32)   -- native fp32 WMMA on gfx1250
__device__ __forceinline__ v8f wmma4(v2f a, v2f b, v8f c) {
  return __builtin_amdgcn_wmma_f32_16x16x4_f32(false, a, false, b, (short)0, c, false, false);
}

__device__ __forceinline__ float sigf(float x) { return 1.0f / (1.0f + __expf(-x)); }

// ---------------------------------------------------------------------------
// Repack W[N][K] (row major) into WMMA B-fragment order so the GEMM inner loop
// is one coalesced b64 load per fragment:
//   WP[((ntile*(K/4)+kstep)*32 + lane)*2 + j] =
//       W[ntile*16 + lane%16][kstep*4 + (lane<16?0:2) + j]
// ---------------------------------------------------------------------------
__global__ void repack_kernel(const float* __restrict__ W, float* __restrict__ WP,
                              int N, int K) {
  int idx = blockIdx.x * 256 + threadIdx.x;       // one float2 per thread
  int total = (N * K) >> 1;
  if (idx >= total) return;
  int lane  = idx & 31;
  int rest  = idx >> 5;
  int ks    = K >> 2;                              // k-steps
  int kstep = rest % ks;
  int ntile = rest / ks;
  int n = ntile * 16 + (lane & 15);
  int k = kstep * 4 + ((lane < 16) ? 0 : 2);
  WP[(size_t)idx * 2 + 0] = W[(size_t)n * K + k + 0];
  WP[(size_t)idx * 2 + 1] = W[(size_t)n * K + k + 1];
}

// ---------------------------------------------------------------------------
// Big input-projection GEMM:  Out[m][n] = sum_k A[m][k]*W[n][k] + bias(n)
// A = x [B*T, DIN] (optionally per-length time-reversed rows), K = DIN = 512.
// Block = 256 thr (8 waves). blockIdx.x = m-tile (16 rows, never crosses a
// batch since T%16==0). blockIdx.y selects a chunk of 8 n-tiles; wave w owns
// n-tile (blockIdx.y*8 + w).  A tile staged in LDS with stride 514 (bank-
// conflict-free b64 fragment reads).  N / reverse are template params so the
// store epilogue folds row strides into immediate offsets, and the K loop
// uses 4 independent accumulators to break the WMMA D->C serial chain.
// ---------------------------------------------------------------------------
template <int N, int REVERSE>
__global__ __launch_bounds__(256)
void gemm_kernel(const float* __restrict__ X, const float* __restrict__ WP,
                 const float* __restrict__ bias0, const float* __restrict__ bias1,
                 float* __restrict__ Out, const int* __restrict__ lens) {
  __shared__ float Atile[16 * 514];
  const int tid = threadIdx.x;
  const int m0  = blockIdx.x * 16;
  const int b   = m0 / T_;                         // whole tile in one batch
  const int len = lens[b];

  // Stage A tile (16 x 512), applying per-length time reversal if requested.
  for (int i = tid; i < 16 * 512; i += 256) {
    int r = i >> 9, c = i & 511;
    int t = (m0 + r) & (T_ - 1);
    int ts = REVERSE ? ((t < len) ? (len - 1 - t) : t) : t;
    Atile[r * 514 + c] = X[((size_t)b * T_ + ts) * DIN_ + c];
  }
  __syncthreads();

  const int wave = tid >> 5, lane = tid & 31;
  const int l16  = lane & 15;
  const int kh   = (lane < 16) ? 0 : 2;
  const int nt   = blockIdx.y * 8 + wave;          // n-tile
  const int n    = nt * 16 + l16;

  v8f acc0 = {}, acc1 = {}, acc2 = {}, acc3 = {};
  const float* wp = WP + (size_t)nt * (DIN_ / 4) * 64 + lane * 2;
  const int arow  = l16 * 514 + kh;
  for (int kk = 0; kk < DIN_ / 4; kk += 4) {
    v2f a0 = *(const v2f*)(&Atile[arow + (kk + 0) * 4]);
    v2f a1 = *(const v2f*)(&Atile[arow + (kk + 1) * 4]);
    v2f a2 = *(const v2f*)(&Atile[arow + (kk + 2) * 4]);
    v2f a3 = *(const v2f*)(&Atile[arow + (kk + 3) * 4]);
    v2f b0 = *(const v2f*)(wp + (kk + 0) * 64);
    v2f b1 = *(const v2f*)(wp + (kk + 1) * 64);
    v2f b2 = *(const v2f*)(wp + (kk + 2) * 64);
    v2f b3 = *(const v2f*)(wp + (kk + 3) * 64);
    acc0 = wmma4(a0, b0, acc0);                    // 4 independent chains
    acc1 = wmma4(a1, b1, acc1);
    acc2 = wmma4(a2, b2, acc2);
    acc3 = wmma4(a3, b3, acc3);
  }
  v8f acc = (acc0 + acc1) + (acc2 + acc3);

  float bias = bias0[n] + (bias1 ? bias1[n] : 0.0f);
  const int rbase = m0 + ((lane < 16) ? 0 : 8);
  float* op = Out + (size_t)rbase * N + n;         // constexpr N -> imm offsets
#pragma unroll
  for (int r = 0; r < 8; ++r)
    op[r * N] = acc[r] + bias;
}

// ---------------------------------------------------------------------------
// Sequential LSTM scan. One workgroup per direction (blockIdx.x = dir),
// 1024 threads = 32 waves. Per step:
//   gates[32][1024] = h[32][256] @ Whh^T   via WMMA (wave w: n-tiles {2w,2w+1},
//   m-tiles {0,1}; 64 K-steps; Whh pre-packed into fragment order),
//   then elementwise gate math with per-thread resident cell state c.
// h lives in LDS (stride 258 -> conflict-free b64 frag reads);
// gates in LDS (stride 1028 -> lane halves hit disjoint banks on C scatter).
// Per-lane global_prefetch of this step's xg rows is issued before the GEMM
// so the ~256-WMMA recurrent matmul hides the HBM latency of the gate stream.
// ---------------------------------------------------------------------------
__global__ __launch_bounds__(1024)
void lstm_scan(const float* __restrict__ xg_f, const float* __restrict__ xg_b,
               const float* __restrict__ WPf, const float* __restrict__ WPb,
               const int* __restrict__ lens, float* __restrict__ lstm_out) {
  const int dir = blockIdx.x;
  const float* xg = dir ? xg_b : xg_f;
  const float* WP = dir ? WPb : WPf;

  extern __shared__ float smem[];
  float* g_lds = smem;                       // 32 * 1028
  float* h_lds = smem + 32 * 1028;           // 32 * 258

  const int tid = threadIdx.x;
  for (int i = tid; i < 32 * 258; i += 1024) h_lds[i] = 0.0f;

  const int wave = tid >> 5, lane = tid & 31;
  const int l16  = lane & 15;
  const int kh   = (lane < 16) ? 0 : 2;
  const int nt0  = wave * 2;
  const float* wp0 = WP + (size_t)nt0 * 64 * 64 + lane * 2;  // 64 ksteps * 64 floats
  const float* wp1 = wp0 + 64 * 64;
  const int ar0 = l16 * 258 + kh;
  const int ar1 = (16 + l16) * 258 + kh;

  // Elementwise ownership: thread -> (b = u*4 + tid>>8, j = tid&255), u=0..7
  const int eb = tid >> 8;
  const int ej = tid & 255;
  float cst[8];
  int   mylen[8];
#pragma unroll
  for (int u = 0; u < 8; ++u) { cst[u] = 0.0f; mylen[u] = lens[u * 4 + eb]; }

  __syncthreads();

  for (int t = 0; t < T_; ++t) {
    // ---- prefetch this step's gate stream (covered by the GEMM below) -----
#pragma unroll
    for (int u = 0; u < 8; ++u) {
      const float* xr = xg + ((size_t)(u * 4 + eb) * T_ + t) * (4 * H_) + ej;
      __builtin_prefetch(xr, 0, 3);
      __builtin_prefetch(xr + H_, 0, 3);
      __builtin_prefetch(xr + 2 * H_, 0, 3);
      __builtin_prefetch(xr + 3 * H_, 0, 3);
    }

    // ---- recurrent GEMM: gates += h @ Whh^T -------------------------------
    v8f a00 = {}, a01 = {}, a10 = {}, a11 = {};
#pragma unroll 4
    for (int kk = 0; kk < 64; ++kk) {
      v2f ha = *(const v2f*)(&h_lds[ar0 + kk * 4]);
      v2f hb = *(const v2f*)(&h_lds[ar1 + kk * 4]);
      v2f b0 = *(const v2f*)(wp0 + kk * 64);
      v2f b1 = *(const v2f*)(wp1 + kk * 64);
      a00 = wmma4(ha, b0, a00);
      a01 = wmma4(ha, b1, a01);
      a10 = wmma4(hb, b0, a10);
      a11 = wmma4(hb, b1, a11);
    }
    const int rb = (lane < 16) ? 0 : 8;
    const int c0 = nt0 * 16 + l16, c1 = c0 + 16;
#pragma unroll
    for (int r = 0; r < 8; ++r) {
      g_lds[(rb + r) * 1028 + c0]      = a00[r];
      g_lds[(rb + r) * 1028 + c1]      = a01[r];
      g_lds[(16 + rb + r) * 1028 + c0] = a10[r];
      g_lds[(16 + rb + r) * 1028 + c1] = a11[r];
    }
    __syncthreads();

    // ---- elementwise gate math -------------------------------------------
#pragma unroll
    for (int u = 0; u < 8; ++u) {
      const int bb = u * 4 + eb;
      const float* xr = xg + ((size_t)bb * T_ + t) * (4 * H_);
      float gi = g_lds[bb * 1028 + ej]           + xr[ej];
      float gf = g_lds[bb * 1028 + H_ + ej]      + xr[H_ + ej];
      float gg = g_lds[bb * 1028 + 2 * H_ + ej]  + xr[2 * H_ + ej];
      float go = g_lds[bb * 1028 + 3 * H_ + ej]  + xr[3 * H_ + ej];
      if (t < mylen[u]) {
        float iv = sigf(gi), fv = sigf(gf), gv = tanhf(gg), ov = sigf(go);
        float cn = fv * cst[u] + iv * gv;
        cst[u] = cn;
        float hn = ov * tanhf(cn);
        h_lds[bb * 258 + ej] = hn;               // masked lanes keep old h
        int tout = dir ? (mylen[u] - 1 - t) : t;
        lstm_out[((size_t)bb * T_ + tout) * (2 * H_) + dir * H_ + ej] = hn;
      }
    }
    __syncthreads();
  }
}

// ---------------------------------------------------------------------------
// Highway combine: flow = sig(g)*lstm + g*(1-sig(g)), masked by length.
// ---------------------------------------------------------------------------
__global__ __launch_bounds__(256)
void highway_kernel(const float* __restrict__ gpre, const float* __restrict__ lstm,
                    const int* __restrict__ lens, float* __restrict__ out) {
  int idx = blockIdx.x * 256 + threadIdx.x;
  if (idx >= B_ * T_ * 2 * H_) return;
  int t = (idx >> 9) & (T_ - 1);
  int b = idx >> 18;
  float gp = gpre[idx];
  float tg = sigf(gp);
  float fl = tg * lstm[idx] + gp * (1.0f - tg);
  out[idx] = (t < lens[b]) ? fl : 0.0f;
}

// ---------------------------------------------------------------------------
extern "C" void kernel_launch(void* const* d_in, const int* in_sizes, int n_in,
                              void* d_out, int out_size, void* d_ws, size_t ws_size,
                              hipStream_t stream) {
  (void)in_sizes; (void)n_in; (void)out_size; (void)ws_size;
  const float* x     = (const float*)d_in[0];
  const float* Wih_f = (const float*)d_in[1];
  const float* Whh_f = (const float*)d_in[2];
  const float* bih_f = (const float*)d_in[3];
  const float* bhh_f = (const float*)d_in[4];
  const float* Wih_b = (const float*)d_in[5];
  const float* Whh_b = (const float*)d_in[6];
  const float* bih_b = (const float*)d_in[7];
  const float* bhh_b = (const float*)d_in[8];
  const float* Wg    = (const float*)d_in[9];
  const float* bg    = (const float*)d_in[10];
  const int*   lens  = (const int*)d_in[11];

  float* ws = (float*)d_ws;
  size_t o = 0;
  float* xg_f   = ws + o;  o += (size_t)B_ * T_ * 4 * H_;   // 16.8M floats
  float* xg_b   = ws + o;  o += (size_t)B_ * T_ * 4 * H_;
  float* g_pre  = ws + o;  o += (size_t)B_ * T_ * 2 * H_;
  float* lstm_o = ws + o;  o += (size_t)B_ * T_ * 2 * H_;
  float* WPf    = ws + o;  o += (size_t)4 * H_ * DIN_;
  float* WPb    = ws + o;  o += (size_t)4 * H_ * DIN_;
  float* WPg    = ws + o;  o += (size_t)2 * H_ * DIN_;
  float* WPhf   = ws + o;  o += (size_t)4 * H_ * H_;
  float* WPhb   = ws + o;  o += (size_t)4 * H_ * H_;

  // 1) repack weights into WMMA B-fragment order
  repack_kernel<<<(4 * H_ * DIN_ / 2 + 255) / 256, 256, 0, stream>>>(Wih_f, WPf, 4 * H_, DIN_);
  repack_kernel<<<(4 * H_ * DIN_ / 2 + 255) / 256, 256, 0, stream>>>(Wih_b, WPb, 4 * H_, DIN_);
  repack_kernel<<<(2 * H_ * DIN_ / 2 + 255) / 256, 256, 0, stream>>>(Wg,    WPg, 2 * H_, DIN_);
  repack_kernel<<<(4 * H_ * H_   / 2 + 255) / 256, 256, 0, stream>>>(Whh_f, WPhf, 4 * H_, H_);
  repack_kernel<<<(4 * H_ * H_   / 2 + 255) / 256, 256, 0, stream>>>(Whh_b, WPhb, 4 * H_, H_);

  // 2) input projections (WMMA GEMMs); backward uses per-length reversed rows
  dim3 g4(B_ * T_ / 16, (4 * H_) / 128);
  gemm_kernel<4 * H_, 0><<<g4, 256, 0, stream>>>(x, WPf, bih_f, bhh_f, xg_f, lens);
  gemm_kernel<4 * H_, 1><<<g4, 256, 0, stream>>>(x, WPb, bih_b, bhh_b, xg_b, lens);
  dim3 g2(B_ * T_ / 16, (2 * H_) / 128);
  gemm_kernel<2 * H_, 0><<<g2, 256, 0, stream>>>(x, WPg, bg, nullptr, g_pre, lens);

  // 3) the two sequential scans in parallel (one WGP each)
  size_t smem = (size_t)(32 * 1028 + 32 * 258) * sizeof(float);  // ~161 KB
  lstm_scan<<<2, 1024, smem, stream>>>(xg_f, xg_b, WPhf, WPhb, lens, lstm_o);

  // 4) highway gate + mask
  highway_kernel<<<(B_ * T_ * 2 * H_ + 255) / 256, 256, 0, stream>>>(g_pre, lstm_o, lens, (float*)d_out);
}